// loss_func2_69621419868219
// MI455X (gfx1250) — compile-verified
//
#include <hip/hip_runtime.h>
#include <hip/hip_bf16.h>

// ---------------- problem constants (fixed by reference setup_inputs) ---------------
constexpr int NN = 8, CC = 8, HH = 1024, WW = 1024;
constexpr int TX = 64, TY = 16;           // spatial tile per block
constexpr int XT = WW / TX;               // 16 tiles in x
constexpr int YT = HH / TY;               // 64 tiles in y
constexpr int NBLK = NN * YT * XT;        // 8192 per-block partials

// cv2.getGaussianKernel(5, 2) taps
constexpr float G0 = 0.15246914f;
constexpr float G1 = 0.22184131f;
constexpr float G2 = 0.25137910f;
constexpr float WL = 1.0f / 67108864.0f;  // 1/(N*C*H*W)
constexpr float WP = 1.0f / 8388608.0f;   // 1/(N*1*H*W)

// ---------------- TDM availability -------------------------------------------------
#ifndef HAVE_TDM
#if defined(__has_builtin)
#if __has_builtin(__builtin_amdgcn_tensor_load_to_lds)
#define HAVE_TDM 1
#endif
#endif
#endif
#ifndef HAVE_TDM
#define HAVE_TDM 0
#endif

typedef __attribute__((ext_vector_type(4))) unsigned int u32x4;
typedef __attribute__((ext_vector_type(4))) int i32x4;
typedef __attribute__((ext_vector_type(8))) int i32x8;

__device__ __forceinline__ int refl101(int i, int n) {
  i = i < 0 ? -i : i;                 // reflect without edge repeat (BORDER_REFLECT_101)
  return i >= n ? 2 * n - 2 - i : i;
}

#if HAVE_TDM
// Issue one TDM descriptor: strided (up to 3-D) global tile -> packed LDS tile.
// tile_z == 0 means 2-D tile. Strides are in elements (data_size = 4 bytes).
// tensor_dim0/1/2 are set to 2^30 (interior tiles never trigger the OOB-zero path).
__device__ __forceinline__ void tdm_tile_load(unsigned lds_addr, const float* gptr,
                                              int tile_x, int tile_y, int tile_z,
                                              int stride0, int stride1) {
  const unsigned long long ga = (unsigned long long)(size_t)gptr;
  u32x4 g0 = {1u,                                  // count=1, user descriptor
              lds_addr,                            // LDS byte offset
              (unsigned)ga,                        // global addr lo
              ((unsigned)(ga >> 32) & 0x01FFFFFFu) | 0x80000000u};  // hi | type=2
  i32x8 g1 = {0x00020000,                          // data_size = 4B
              0,                                   // tensor_dim0[15:0] (2^30 -> 0)
              0x00004000,                          // tensor_dim0[31:16] | tdim1 lo
              0x00004000 | (tile_x << 16),         // tensor_dim1[31:16] | tile_dim0
              tile_y | (tile_z << 16),             // tile_dim1 | tile_dim2
              stride0,                             // tensor_dim0_stride[31:0]
              (stride1 & 0xFFFF) << 16,            // stride0 hi16 | stride1[15:0]
              stride1 >> 16};                      // tensor_dim1_stride[47:16]
  i32x4 g2 = {1 << 30, 0, 0, 0};                   // tensor_dim2 (unused if tile_z==0)
  i32x4 g3 = {0, 0, 0, 0};
#if __clang_major__ >= 23
  i32x8 gz = (i32x8)0;
  __builtin_amdgcn_tensor_load_to_lds(g0, g1, g2, g3, gz, 0);
#else
  __builtin_amdgcn_tensor_load_to_lds(g0, g1, g2, g3, 0);
#endif
}
#endif

// =====================================================================================
// One block per (n, 16x64 tile). Produces one weighted partial-loss float in ws[].
// =====================================================================================
__global__ __launch_bounds__(256) void fused_pan_loss(
    const float* __restrict__ refp, const float* __restrict__ panp,
    const float* __restrict__ msp,  const float* __restrict__ outp,
    float* __restrict__ ws) {
  __shared__ float s_out[CC][TY + 4][TX + 4];   // raw `out` tile, blur halo 2
  __shared__ float s_hb[2][TY + 4][TX];         // horizontal-blur intermediate, 2 ch
  __shared__ float s_pan[TY + 2][TX + 2];       // pan tile, zero-padded halo 1
  __shared__ float s_o2p[TY + 2][TX + 2];       // channel-mean of out, zero-padded halo 1
  __shared__ float s_red[8];

  const int tid = threadIdx.x;
  const int xt = blockIdx.x, yt = blockIdx.y, n = blockIdx.z;
  const int x0 = xt * TX, y0 = yt * TY;
  const long nbase = (long)n * CC * HH * WW;
  const long pbase = (long)n * HH * WW;

  // ---- tile fill: interior -> two TDM descriptors; border -> cooperative loads ----
  bool coop = true;
#if HAVE_TDM
  const bool interior =
      (x0 >= 2) && (x0 + TX + 2 <= WW) && (y0 >= 2) && (y0 + TY + 2 <= HH);
  if (interior) {
    coop = false;
    if (tid < 32) {  // one wave issues the DMAs (TDM ignores EXEC; per-wave issue)
      tdm_tile_load((unsigned)(unsigned long long)(void*)&s_out[0][0][0],
                    outp + nbase + (long)(y0 - 2) * WW + (x0 - 2),
                    TX + 4, TY + 4, CC, WW, HH * WW);          // 68 x 20 x 8
      tdm_tile_load((unsigned)(unsigned long long)(void*)&s_pan[0][0],
                    panp + pbase + (long)(y0 - 1) * WW + (x0 - 1),
                    TX + 2, TY + 2, 0, WW, 0);                 // 66 x 18 (2-D)
      __builtin_amdgcn_s_wait_tensorcnt(0);                    // both DMAs done
    }
  }
#endif
  if (coop) {
    // pan tile: zero padding matches Sobel's Conv2d(padding=1)
    for (int i = tid; i < (TY + 2) * (TX + 2); i += 256) {
      const int yy = i / (TX + 2), xx = i % (TX + 2);
      const int gy = y0 - 1 + yy, gx = x0 - 1 + xx;
      float v = 0.0f;
      if (gy >= 0 && gy < HH && gx >= 0 && gx < WW)
        v = panp[pbase + (long)gy * WW + gx];
      s_pan[yy][xx] = v;
    }
    // out tile: reflect-101 padding matches cv2.GaussianBlur's default border
    for (int i = tid; i < CC * (TY + 4) * (TX + 4); i += 256) {
      const int c = i / ((TY + 4) * (TX + 4));
      const int r = i % ((TY + 4) * (TX + 4));
      const int yy = r / (TX + 4), xx = r % (TX + 4);
      const int gy = refl101(y0 - 2 + yy, HH), gx = refl101(x0 - 2 + xx, WW);
      s_out[c][yy][xx] = outp[nbase + (long)c * HH * WW + (long)gy * WW + gx];
    }
  }
  __syncthreads();

  // ---- out2pan = mean_c(out), zero-padded halo 1 (valid values exist: halo2 >= halo1)
  for (int i = tid; i < (TY + 2) * (TX + 2); i += 256) {
    const int yy = i / (TX + 2), xx = i % (TX + 2);
    const int gy = y0 - 1 + yy, gx = x0 - 1 + xx;
    float v = 0.0f;
    if (gy >= 0 && gy < HH && gx >= 0 && gx < WW) {
      float s = 0.0f;
#pragma unroll
      for (int c = 0; c < CC; ++c) s += s_out[c][yy + 1][xx + 1];
      v = s * 0.125f;
    }
    s_o2p[yy][xx] = v;
  }
  __syncthreads();

  // ---- per-thread pixel set: x = tid&63, rows y = (tid>>6) + {0,4,8,12} ----
  const int lx = tid & (TX - 1);
  const int ry = tid >> 6;

  // Sobel term: |gx(pan)-gx(out2pan)| + |gy(pan)-gy(out2pan)|
  float accP = 0.0f;
#pragma unroll
  for (int k = 0; k < TY / 4; ++k) {
    const int y = ry + 4 * k;
    const float p00 = s_pan[y][lx],     p01 = s_pan[y][lx + 1],     p02 = s_pan[y][lx + 2];
    const float p10 = s_pan[y + 1][lx],                             p12 = s_pan[y + 1][lx + 2];
    const float p20 = s_pan[y + 2][lx], p21 = s_pan[y + 2][lx + 1], p22 = s_pan[y + 2][lx + 2];
    const float q00 = s_o2p[y][lx],     q01 = s_o2p[y][lx + 1],     q02 = s_o2p[y][lx + 2];
    const float q10 = s_o2p[y + 1][lx],                             q12 = s_o2p[y + 1][lx + 2];
    const float q20 = s_o2p[y + 2][lx], q21 = s_o2p[y + 2][lx + 1], q22 = s_o2p[y + 2][lx + 2];
    const float gxp = (p02 - p00) + 2.0f * (p12 - p10) + (p22 - p20);
    const float gyp = (p20 - p00) + 2.0f * (p21 - p01) + (p22 - p02);
    const float gxq = (q02 - q00) + 2.0f * (q12 - q10) + (q22 - q20);
    const float gyq = (q20 - q00) + 2.0f * (q21 - q01) + (q22 - q02);
    accP += fabsf(gxp - gxq) + fabsf(gyp - gyq);
  }

  // ---- Gaussian blur (separable, 2 channels per pass) + L1 terms ----
  float accMs = 0.0f, accRef = 0.0f;
  for (int c0 = 0; c0 < CC; c0 += 2) {
    // keep the NT stream of the *next* channel pair ahead of the ALUs
    if (c0 + 2 < CC) {
      const long nidx = nbase + (long)(c0 + 2) * (HH * WW) +
                        (long)(y0 + ry) * WW + (x0 + lx);
      __builtin_prefetch(&msp[nidx], 0, 1);
      __builtin_prefetch(&refp[nidx], 0, 1);
    }
    for (int i = tid; i < 2 * (TY + 4) * TX; i += 256) {
      const int cc = i / ((TY + 4) * TX);
      const int r = i % ((TY + 4) * TX);
      const int yy = r / TX, xx = r % TX;
      const float* row = &s_out[c0 + cc][yy][xx];
      s_hb[cc][yy][xx] =
          G0 * (row[0] + row[4]) + G1 * (row[1] + row[3]) + G2 * row[2];
    }
    __syncthreads();
#pragma unroll
    for (int k = 0; k < TY / 4; ++k) {
      const int y = ry + 4 * k;
      const long pix = nbase + (long)(y0 + y) * WW + (x0 + lx);
#pragma unroll
      for (int cc = 0; cc < 2; ++cc) {
        const int c = c0 + cc;
        const float b = G0 * (s_hb[cc][y][lx] + s_hb[cc][y + 4][lx]) +
                        G1 * (s_hb[cc][y + 1][lx] + s_hb[cc][y + 3][lx]) +
                        G2 * s_hb[cc][y + 2][lx];
        const long idx = pix + (long)c * (HH * WW);
        const float mv = __builtin_nontemporal_load(&msp[idx]);   // streamed once
        const float rv = __builtin_nontemporal_load(&refp[idx]);  // streamed once
        const float ov = s_out[c][y + 2][lx + 2];
        accMs  += fabsf(b - mv);
        accRef += fabsf(ov - rv);
      }
    }
    __syncthreads();
  }

  // ---- deterministic block reduction (wave32 shuffle tree + LDS across 8 waves) ----
  float part = (accMs + accRef) * WL + accP * WP;
#pragma unroll
  for (int o = 16; o > 0; o >>= 1) part += __shfl_down(part, o, 32);
  if ((tid & 31) == 0) s_red[tid >> 5] = part;
  __syncthreads();
  if (tid == 0) {
    float t = 0.0f;
#pragma unroll
    for (int w = 0; w < 8; ++w) t += s_red[w];
    ws[((long)n * YT + yt) * XT + xt] = t;
  }
}

// =====================================================================================
// Fixed-order final reduction of 8192 per-block partials -> scalar loss.
// =====================================================================================
__global__ __launch_bounds__(256) void final_reduce(const float* __restrict__ ws,
                                                    float* __restrict__ outv, int n) {
  __shared__ float s_red[8];
  float a = 0.0f;
  for (int i = threadIdx.x; i < n; i += 256) a += ws[i];
#pragma unroll
  for (int o = 16; o > 0; o >>= 1) a += __shfl_down(a, o, 32);
  if ((threadIdx.x & 31) == 0) s_red[threadIdx.x >> 5] = a;
  __syncthreads();
  if (threadIdx.x == 0) {
    float t = 0.0f;
#pragma unroll
    for (int w = 0; w < 8; ++w) t += s_red[w];
    outv[0] = t;
  }
}

extern "C" void kernel_launch(void* const* d_in, const int* in_sizes, int n_in,
                              void* d_out, int out_size, void* d_ws, size_t ws_size,
                              hipStream_t stream) {
  (void)in_sizes; (void)n_in; (void)out_size; (void)ws_size;
  const float* refp = (const float*)d_in[0];
  const float* panp = (const float*)d_in[1];
  const float* msp  = (const float*)d_in[2];
  const float* outp = (const float*)d_in[3];
  float* ws = (float*)d_ws;

  dim3 grid(XT, YT, NN);   // (16, 64, 8) -> 8192 blocks
  dim3 block(256);
  fused_pan_loss<<<grid, block, 0, stream>>>(refp, panp, msp, outp, ws);
  final_reduce<<<1, 256, 0, stream>>>(ws, (float*)d_out, NBLK);
}